// MultiHeadSelfAttention_83262236000353
// MI455X (gfx1250) — compile-verified
//
#include <hip/hip_runtime.h>
#include <math.h>
#include <stdint.h>

// MI455X / gfx1250: wave32, fp32 WMMA 16x16x4 + TDM (tensor_load_to_lds) staging.

typedef __attribute__((ext_vector_type(2))) float v2f;
typedef __attribute__((ext_vector_type(4))) float v4f;
typedef __attribute__((ext_vector_type(8))) float v8f;
typedef __attribute__((ext_vector_type(4))) uint32_t v4u;
typedef __attribute__((ext_vector_type(8))) uint32_t v8u;

#define WMMA_F32(a, b, c) \
  __builtin_amdgcn_wmma_f32_16x16x4_f32(false, (a), false, (b), (short)0, (c), false, false)

// Problem constants (fixed by the reference)
constexpr int Bc = 4, Sc = 2048, Ec = 1024, Hc = 16, Dc = 64;
constexpr int Mc = Bc * Sc;      // 8192 tokens
constexpr int LDP = 20;          // padded LDS row stride (floats): bank-conflict-free, 16B aligned

// ---------------------------------------------------------------------------
// TDM: DMA a 2-D tile [rows x 16 floats] (row stride = 1024 floats in memory)
// into LDS with hardware padding 16DW + 4DW -> LDS row stride 20 floats.
// D# packing per CDNA5 ISA 8.3/8.4. 2-D tile -> groups 0,1 only (VADDR2/3 NULL).
// Wave-level op (EXEC ignored); caller guards so only one wave issues it.
// ---------------------------------------------------------------------------
__device__ __forceinline__ void tdm_load_tile(const float* gsrc, unsigned lds_byte,
                                              unsigned rows) {
  const uint64_t ga = (uint64_t)(uintptr_t)gsrc;
  v4u g0;
  g0[0] = 1u;                                   // count=1, is_restore=0, no gather
  g0[1] = lds_byte;                             // lds_addr (bytes)
  g0[2] = (uint32_t)ga;                         // global_addr[31:0]
  g0[3] = (uint32_t)(ga >> 32) | (2u << 30);    // global_addr[56:32] | type=2
  v8u g1;
  g1[0] = (2u << 16)        // data_size = 4 bytes
        | (1u << 20)        // pad_enable
        | (3u << 22)        // pad_interval: 16 DWORDs
        | (3u << 25);       // pad_amount:   4 DWORDs  -> LDS stride 20 floats
  g1[1] = (1024u & 0xFFFFu) << 16;                        // tensor_dim0[15:0]
  g1[2] = (1024u >> 16) | ((8192u & 0xFFFFu) << 16);      // td0[31:16] | td1[15:0]
  g1[3] = (8192u >> 16) | (16u << 16);                    // td1[31:16] | tile_dim0=16
  g1[4] = rows;                                           // tile_dim1 | tile_dim2=0
  g1[5] = 1024u;                                          // tensor_dim0_stride[31:0]
  g1[6] = 0u;                                             // stride hi | dim1_stride lo
  g1[7] = 0u;
  asm volatile("tensor_load_to_lds %0, %1" :: "s"(g0), "s"(g1) : "memory");
}

// ---------------------------------------------------------------------------
// GEMM: C[m,n] = sum_k A[m,k] * W[n,k] + bias[n]   (y = A @ W^T + b)
// A row-major [Mc x Ec], W row-major [Ec x Ec].
// mode 0: store row-major [Mc x Ec];  mode 1: store split-heads [B,H,S,D].
// Block tile 128(M) x 64(N), BK=16, 256 threads = 8 waves, wave = 2x2 WMMA tiles.
// Double-buffered LDS fed by TDM; wave 0 is the DMA producer.
// ---------------------------------------------------------------------------
__global__ __launch_bounds__(256) void gemm_xwt(const float* __restrict__ A,
                                                const float* __restrict__ W,
                                                const float* __restrict__ bias,
                                                float* __restrict__ Cout,
                                                int mode) {
  __shared__ float As[2][128 * LDP];
  __shared__ float Ws[2][64 * LDP];

  const int t = threadIdx.x;
  const int lane = t & 31, w = t >> 5;
  const int m_base = blockIdx.y * 128;
  const int n_base = blockIdx.x * 64;
  const int wm = (w >> 1) * 32;   // wave M offset within block tile
  const int wn = (w & 1) * 32;    // wave N offset within block tile
  const int l15 = lane & 15, hi = lane >> 4;

  const float* Ablk = A + (size_t)m_base * Ec;
  const float* Wblk = W + (size_t)n_base * Ec;

  v8f acc[2][2] = {};

  // Prologue: DMA stage 0 into buffer 0
  if (w == 0) {
    tdm_load_tile(Ablk, (unsigned)(uintptr_t)&As[0][0], 128u);
    tdm_load_tile(Wblk, (unsigned)(uintptr_t)&Ws[0][0], 64u);
  }

  int cur = 0;
  for (int k0 = 0; k0 < Ec; k0 += 16) {
    if (w == 0) {
      if (k0 + 16 < Ec) {
        // issue next stage into the other buffer, then wait for current stage
        tdm_load_tile(Ablk + (k0 + 16), (unsigned)(uintptr_t)&As[cur ^ 1][0], 128u);
        tdm_load_tile(Wblk + (k0 + 16), (unsigned)(uintptr_t)&Ws[cur ^ 1][0], 64u);
        __builtin_amdgcn_s_wait_tensorcnt(2);   // current stage's 2 loads complete
      } else {
        __builtin_amdgcn_s_wait_tensorcnt(0);
      }
    }
    __syncthreads();   // release: buf[cur] is valid for all waves

    const float* Asb = &As[cur][0];
    const float* Wsb = &Ws[cur][0];
#pragma unroll
    for (int s = 0; s < 4; ++s) {
      const int col = s * 4 + (hi << 1);   // K pair per ISA A/B 16x4 layout
      v2f af0 = *(const v2f*)&Asb[(wm + l15) * LDP + col];
      v2f af1 = *(const v2f*)&Asb[(wm + 16 + l15) * LDP + col];
      v2f bf0 = *(const v2f*)&Wsb[(wn + l15) * LDP + col];
      v2f bf1 = *(const v2f*)&Wsb[(wn + 16 + l15) * LDP + col];
      acc[0][0] = WMMA_F32(af0, bf0, acc[0][0]);
      acc[0][1] = WMMA_F32(af0, bf1, acc[0][1]);
      acc[1][0] = WMMA_F32(af1, bf0, acc[1][0]);
      acc[1][1] = WMMA_F32(af1, bf1, acc[1][1]);
    }
    __syncthreads();   // all reads of buf[cur] done before producer overwrites it
    cur ^= 1;
  }

  // Store (C/D layout: VGPR r -> row r (lanes 0-15) / r+8 (lanes 16-31); lane&15 -> col)
#pragma unroll
  for (int mi = 0; mi < 2; ++mi)
#pragma unroll
    for (int ni = 0; ni < 2; ++ni) {
      const int n = n_base + wn + ni * 16 + l15;
      const float bv = bias[n];
#pragma unroll
      for (int r = 0; r < 8; ++r) {
        const int m = m_base + wm + mi * 16 + r + (hi << 3);
        const float val = acc[mi][ni][r] + bv;
        size_t off;
        if (mode == 0) {
          off = (size_t)m * Ec + n;
        } else {
          const int b = m >> 11, s2 = m & (Sc - 1);
          const int h = n >> 6, d = n & (Dc - 1);
          off = ((((size_t)b * Hc + h) << 11) + s2) * Dc + d;  // [B,H,S,D]
        }
        Cout[off] = val;
      }
    }
}

// ---------------------------------------------------------------------------
// Wave-level flash attention (causal, fp32 WMMA).
// One wave = one 16-row query tile of one (b,h). Q/K/V in [B,H,S,D].
// Output written pre-concatenated [B,S,E] to scratch.
// ---------------------------------------------------------------------------
__global__ __launch_bounds__(256) void flash_attn(const float* __restrict__ Q,
                                                  const float* __restrict__ K,
                                                  const float* __restrict__ V,
                                                  float* __restrict__ O) {
  __shared__ float Pst[8 * 16 * LDP];  // per-wave 16x20 P staging

  const int t = threadIdx.x, lane = t & 31, w = t >> 5;
  float* Pw = &Pst[w * 16 * LDP];

  const int gw = blockIdx.x * 8 + w;   // 8192 waves total
  const int qt = gw & 127;             // 128 q-tiles per (b,h)
  const int bh = gw >> 7;              // 0..63
  const int b = bh >> 4, h = bh & 15;
  const int q0 = qt << 4;
  const int l15 = lane & 15, hi = lane >> 4;
  const int mhalf = hi << 3;

  const float* Qb = Q + (size_t)bh * Sc * Dc;
  const float* Kb = K + (size_t)bh * Sc * Dc;
  const float* Vb = V + (size_t)bh * Sc * Dc;

  // Preload Q A-fragments for the full head dim (16 K-steps of 4)
  v2f aq[16];
#pragma unroll
  for (int kk = 0; kk < 16; ++kk) {
    const int col = kk * 4 + (hi << 1);
    aq[kk] = *(const v2f*)&Qb[(size_t)(q0 + l15) * Dc + col];
  }

  v8f o0 = {}, o1 = {}, o2 = {}, o3 = {};
  float mi[8], li[8];
#pragma unroll
  for (int r = 0; r < 8; ++r) { mi[r] = -INFINITY; li[r] = 0.f; }

  for (int j0 = 0; j0 <= q0; j0 += 16) {
    // S = Q K^T (16x16), K-dim = 64
    v8f s = {};
#pragma unroll
    for (int kk = 0; kk < 16; ++kk) {
      const int col = kk * 4 + (hi << 1);
      v2f bk = *(const v2f*)&Kb[(size_t)(j0 + l15) * Dc + col];
      s = WMMA_F32(aq[kk], bk, s);
    }

    // scale, causal mask, online softmax (row = r + mhalf, col = l15;
    // each row lives in one 16-lane half -> shfl_xor width 16)
    const float scale = 0.125f;  // 1/sqrt(64)
#pragma unroll
    for (int r = 0; r < 8; ++r) {
      float sv = s[r] * scale;
      const int m = r + mhalf;
      if (j0 + l15 > q0 + m) sv = -INFINITY;

      float rm = sv;
      rm = fmaxf(rm, __shfl_xor(rm, 1, 16));
      rm = fmaxf(rm, __shfl_xor(rm, 2, 16));
      rm = fmaxf(rm, __shfl_xor(rm, 4, 16));
      rm = fmaxf(rm, __shfl_xor(rm, 8, 16));

      const float mnew = fmaxf(mi[r], rm);
      const float corr = __expf(mi[r] - mnew);
      const float pv = __expf(sv - mnew);

      float rs = pv;
      rs += __shfl_xor(rs, 1, 16);
      rs += __shfl_xor(rs, 2, 16);
      rs += __shfl_xor(rs, 4, 16);
      rs += __shfl_xor(rs, 8, 16);

      li[r] = li[r] * corr + rs;
      mi[r] = mnew;
      o0[r] *= corr; o1[r] *= corr; o2[r] *= corr; o3[r] *= corr;

      Pw[m * LDP + l15] = pv;  // stage P (C layout -> LDS tile)
    }
    // same-wave LDS RAW ordering, explicit CDNA5 DS counter wait
    asm volatile("s_wait_dscnt 0x0" ::: "memory");

    // Re-stripe P as A-fragments
    v2f pa[4];
#pragma unroll
    for (int s2 = 0; s2 < 4; ++s2) {
      const int col = s2 * 4 + (hi << 1);
      pa[s2] = *(const v2f*)&Pw[l15 * LDP + col];
    }

    // O += P @ V_block  (M=16, K=16, N=64)
#pragma unroll
    for (int s2 = 0; s2 < 4; ++s2) {
      const int krow = j0 + s2 * 4 + (hi << 1);
      const float* vrow0 = &Vb[(size_t)krow * Dc + l15];
      v2f bv;
      bv.x = vrow0[0];        bv.y = vrow0[Dc];        o0 = WMMA_F32(pa[s2], bv, o0);
      bv.x = vrow0[16];       bv.y = vrow0[Dc + 16];   o1 = WMMA_F32(pa[s2], bv, o1);
      bv.x = vrow0[32];       bv.y = vrow0[Dc + 32];   o2 = WMMA_F32(pa[s2], bv, o2);
      bv.x = vrow0[48];       bv.y = vrow0[Dc + 48];   o3 = WMMA_F32(pa[s2], bv, o3);
    }
  }

  // Normalize + store concatenated heads: O[b, s, h*64 + d]
#pragma unroll
  for (int r = 0; r < 8; ++r) {
    const int m = q0 + r + mhalf;
    const float inv = 1.0f / li[r];
    const size_t base = ((size_t)b * Sc + m) * Ec + h * Dc + l15;
    O[base + 0]  = o0[r] * inv;
    O[base + 16] = o1[r] * inv;
    O[base + 32] = o2[r] * inv;
    O[base + 48] = o3[r] * inv;
  }
}

// ---------------------------------------------------------------------------
extern "C" void kernel_launch(void* const* d_in, const int* in_sizes, int n_in,
                              void* d_out, int out_size, void* d_ws, size_t ws_size,
                              hipStream_t stream) {
  (void)in_sizes; (void)n_in; (void)out_size; (void)ws_size;

  const float* x  = (const float*)d_in[0];
  const float* Wq = (const float*)d_in[1];
  const float* bq = (const float*)d_in[2];
  const float* Wk = (const float*)d_in[3];
  const float* bk = (const float*)d_in[4];
  const float* Wv = (const float*)d_in[5];
  const float* bv = (const float*)d_in[6];
  const float* Wo = (const float*)d_in[7];
  const float* bo = (const float*)d_in[8];

  float* out = (float*)d_out;
  const size_t nBSE = (size_t)Bc * Sc * Ec;   // == B*H*S*D
  float* Kout = out + nBSE;                   // d_out: [out | k | v]
  float* Vout = Kout + nBSE;

  float* Qbuf = (float*)d_ws;                 // [B,H,S,D]
  float* Abuf = Qbuf + nBSE;                  // attention output, [B,S,E]

  dim3 gblk(Ec / 64, Mc / 128);               // (16, 64)

  gemm_xwt<<<gblk, 256, 0, stream>>>(x, Wq, bq, Qbuf, 1);
  gemm_xwt<<<gblk, 256, 0, stream>>>(x, Wk, bk, Kout, 1);
  gemm_xwt<<<gblk, 256, 0, stream>>>(x, Wv, bv, Vout, 1);

  flash_attn<<<(Bc * Hc * (Sc / 16)) / 8, 256, 0, stream>>>(Qbuf, Kout, Vout, Abuf);

  gemm_xwt<<<gblk, 256, 0, stream>>>(Abuf, Wo, bo, out, 0);
}